// Net_64982855188860
// MI455X (gfx1250) — compile-verified
//
#include <hip/hip_runtime.h>

#define N_NODES   100000
#define N_EDGES   1000000
#define F_IN      8
#define S_EDGE    4
#define CHANNELS  32
#define N_GRAPHS  64
#define N_LABELS  10
#define NTILES    (N_EDGES / 16)   // 62500 exact

typedef __attribute__((ext_vector_type(2))) float v2f;
typedef __attribute__((ext_vector_type(4))) float v4f;
typedef __attribute__((ext_vector_type(8))) float v8f;

static __device__ __forceinline__ v2f mk2(float a, float b) { v2f r; r.x = a; r.y = b; return r; }

// ---------------------------------------------------------------------------
// 0) zero the accumulation workspace (agg [N,32] + pooled [G,32], contiguous)
// ---------------------------------------------------------------------------
__global__ void zero_ws_kernel(float* __restrict__ p, int n) {
  int i = blockIdx.x * blockDim.x + threadIdx.x;
  int stride = gridDim.x * blockDim.x;
  for (; i < n; i += stride) p[i] = 0.0f;
}

// ---------------------------------------------------------------------------
// 1) Edge GEMM via V_WMMA_F32_16X16X4_F32:
//    per 16-edge tile: D[16x32] = Z[16x40] @ W40[40x32], scatter-add into agg.
//    Z[k] : k<32 -> e[k>>3]*x[k&7] ; k>=32 -> x[k-32]  (bias rows)
//    W40[k][c]: k<32 -> fgn_w[(k>>3)*256 + (k&7)*32 + c] ; else fgn_b[(k-32)*32+c]
// ---------------------------------------------------------------------------
__global__ void __launch_bounds__(256)
ecc_edge_wmma_kernel(const float* __restrict__ x, const int* __restrict__ src,
                     const int* __restrict__ dst, const float* __restrict__ e,
                     const float* __restrict__ fgn_w, const float* __restrict__ fgn_b,
                     float* __restrict__ agg)
{
  const unsigned lane = threadIdx.x & 31u;
  const unsigned l16  = lane & 15u;
  const unsigned half = lane >> 4;            // selects K pair {2h, 2h+1} within each K4 group
  const int wavesPerBlock = blockDim.x >> 5;
  const int wave  = blockIdx.x * wavesPerBlock + (threadIdx.x >> 5);
  const int nWave = gridDim.x * wavesPerBlock;

  // --- loop-invariant B fragments: 10 K-groups x 2 N-tiles, kept in VGPRs ---
  // B(4x16) layout (mirrors documented A 16x4): VGPR0 = row K=2h, VGPR1 = K=2h+1
  v2f bf[10][2];
#pragma unroll
  for (int g = 0; g < 10; ++g) {
    const unsigned k0 = 4u * g + 2u * half;
#pragma unroll
    for (int t = 0; t < 2; ++t) {
      const unsigned c = (unsigned)t * 16u + l16;
      const float* r0;
      if (4 * g < 32) {                        // compile-time per g: fgn_w rows
        const unsigned s = k0 >> 3, fi = k0 & 7u;
        r0 = fgn_w + (s * (F_IN * CHANNELS) + fi * CHANNELS + c);
      } else {                                 // bias rows
        r0 = fgn_b + ((k0 - 32u) * CHANNELS + c);
      }
      bf[g][t].x = r0[0];
      bf[g][t].y = r0[CHANNELS];               // next K row is +32 floats in both tables
    }
  }

  for (int tile = wave; tile < NTILES; tile += nWave) {   // uniform: EXEC stays all-1s
    const unsigned rbase = (unsigned)tile * 16u;
    const unsigned erow  = rbase + l16;                    // both halves mirror rows 0..15

    // prefetch next grid-stride tile (streaming e/src/dst)
    if (tile + nWave < NTILES) {
      const unsigned nrow = (unsigned)(tile + nWave) * 16u + l16;
      __builtin_prefetch(e + nrow * 4u, 0, 0);
      __builtin_prefetch(src + nrow, 0, 0);
      __builtin_prefetch(dst + nrow, 0, 0);
    }

    const unsigned sn = (unsigned)src[erow];
    const v4f xa = *(const v4f*)(x + sn * 8u);
    const v4f xb = *(const v4f*)(x + sn * 8u + 4u);
    const v4f ev = *(const v4f*)(e + erow * 4u);
    const float x0 = xa.x, x1 = xa.y, x2 = xa.z, x3 = xa.w;
    const float x4 = xb.x, x5 = xb.y, x6 = xb.z, x7 = xb.w;
    const float e0 = ev.x, e1 = ev.y, e2 = ev.z, e3 = ev.w;

    // A fragments built in-register (compiler folds the halves via op_sel)
    v2f af[10];
    if (half == 0u) {             // K = 4g, 4g+1
      af[0] = mk2(e0 * x0, e0 * x1);
      af[1] = mk2(e0 * x4, e0 * x5);
      af[2] = mk2(e1 * x0, e1 * x1);
      af[3] = mk2(e1 * x4, e1 * x5);
      af[4] = mk2(e2 * x0, e2 * x1);
      af[5] = mk2(e2 * x4, e2 * x5);
      af[6] = mk2(e3 * x0, e3 * x1);
      af[7] = mk2(e3 * x4, e3 * x5);
      af[8] = mk2(x0, x1);
      af[9] = mk2(x4, x5);
    } else {                      // K = 4g+2, 4g+3
      af[0] = mk2(e0 * x2, e0 * x3);
      af[1] = mk2(e0 * x6, e0 * x7);
      af[2] = mk2(e1 * x2, e1 * x3);
      af[3] = mk2(e1 * x6, e1 * x7);
      af[4] = mk2(e2 * x2, e2 * x3);
      af[5] = mk2(e2 * x6, e2 * x7);
      af[6] = mk2(e3 * x2, e3 * x3);
      af[7] = mk2(e3 * x6, e3 * x7);
      af[8] = mk2(x2, x3);
      af[9] = mk2(x6, x7);
    }

    v8f acc0 = {};   // channels  0..15
    v8f acc1 = {};   // channels 16..31
#pragma unroll
    for (int g = 0; g < 10; ++g) {
      acc0 = __builtin_amdgcn_wmma_f32_16x16x4_f32(false, af[g], false, bf[g][0],
                                                   (short)0, acc0, false, false);
      acc1 = __builtin_amdgcn_wmma_f32_16x16x4_f32(false, af[g], false, bf[g][1],
                                                   (short)0, acc1, false, false);
    }

    // D layout: VGPR v -> row v + 8*half, col = l16 (+16 for acc1).
    // Unsigned 32-bit offsets off the uniform base -> saddr-form atomics.
#pragma unroll
    for (int v = 0; v < 8; ++v) {
      const unsigned row = (unsigned)v + half * 8u;
      const unsigned d   = (unsigned)dst[rbase + row];
      const unsigned idx = (d << 5) + l16;
      unsafeAtomicAdd(&agg[idx],       acc0[v]);
      unsafeAtomicAdd(&agg[idx + 16u], acc1[v]);
    }
  }
}

// ---------------------------------------------------------------------------
// 2) h = relu(agg + x@root_w + conv_b); pooled[g] += h   (i sorted -> run-sum)
//    lane == channel; one atomic per (graph boundary, channel)
// ---------------------------------------------------------------------------
__global__ void __launch_bounds__(256)
node_relu_pool_kernel(const float* __restrict__ x, const float* __restrict__ agg,
                      const float* __restrict__ root_w, const float* __restrict__ conv_b,
                      const int* __restrict__ gidx, float* __restrict__ pooled,
                      int nodesPerWave)
{
  const unsigned lane = threadIdx.x & 31u;  // channel
  const int wave = blockIdx.x * (blockDim.x >> 5) + (threadIdx.x >> 5);
  const int n0 = wave * nodesPerWave;
  if (n0 >= N_NODES) return;
  const int n1 = min(n0 + nodesPerWave, N_NODES);

  float rw0 = root_w[0 * CHANNELS + lane], rw1 = root_w[1 * CHANNELS + lane];
  float rw2 = root_w[2 * CHANNELS + lane], rw3 = root_w[3 * CHANNELS + lane];
  float rw4 = root_w[4 * CHANNELS + lane], rw5 = root_w[5 * CHANNELS + lane];
  float rw6 = root_w[6 * CHANNELS + lane], rw7 = root_w[7 * CHANNELS + lane];
  const float cb = conv_b[lane];

  float acc = 0.0f;
  int cur = gidx[n0];
  for (int n = n0; n < n1; ++n) {
    const int g = gidx[n];
    if (g != cur) {
      unsafeAtomicAdd(&pooled[(unsigned)cur * 32u + lane], acc);
      acc = 0.0f; cur = g;
    }
    const unsigned un = (unsigned)n;
    const v4f xa = *(const v4f*)(x + un * 8u);
    const v4f xb = *(const v4f*)(x + un * 8u + 4u);
    float h = agg[un * 32u + lane] + cb
            + xa.x * rw0 + xa.y * rw1 + xa.z * rw2 + xa.w * rw3
            + xb.x * rw4 + xb.y * rw5 + xb.z * rw6 + xb.w * rw7;
    acc += fmaxf(h, 0.0f);
  }
  unsafeAtomicAdd(&pooled[(unsigned)cur * 32u + lane], acc);
}

// ---------------------------------------------------------------------------
// 3) softmax(pooled @ dense_w + dense_b)  -> out [64 x 10]
// ---------------------------------------------------------------------------
__global__ void head_softmax_kernel(const float* __restrict__ pooled,
                                    const float* __restrict__ dense_w,
                                    const float* __restrict__ dense_b,
                                    float* __restrict__ out)
{
  const int g = blockIdx.x * blockDim.x + threadIdx.x;
  if (g >= N_GRAPHS) return;
  float p[CHANNELS];
#pragma unroll
  for (int k = 0; k < CHANNELS; ++k) p[k] = pooled[(unsigned)g * 32u + (unsigned)k];
  float logit[N_LABELS];
  float m = -3.0e38f;
#pragma unroll
  for (int c = 0; c < N_LABELS; ++c) {
    float s = dense_b[c];
#pragma unroll
    for (int k = 0; k < CHANNELS; ++k) s += p[k] * dense_w[k * N_LABELS + c];
    logit[c] = s;
    m = fmaxf(m, s);
  }
  float sum = 0.0f;
#pragma unroll
  for (int c = 0; c < N_LABELS; ++c) { logit[c] = __expf(logit[c] - m); sum += logit[c]; }
  const float inv = 1.0f / sum;
#pragma unroll
  for (int c = 0; c < N_LABELS; ++c) out[(unsigned)g * N_LABELS + (unsigned)c] = logit[c] * inv;
}

// ---------------------------------------------------------------------------
extern "C" void kernel_launch(void* const* d_in, const int* in_sizes, int n_in,
                              void* d_out, int out_size, void* d_ws, size_t ws_size,
                              hipStream_t stream) {
  (void)in_sizes; (void)n_in; (void)out_size; (void)ws_size;
  const float* x       = (const float*)d_in[0];
  const int*   src     = (const int*)  d_in[1];
  const int*   dst     = (const int*)  d_in[2];
  const float* e       = (const float*)d_in[3];
  const int*   gidx    = (const int*)  d_in[4];
  const float* fgn_w   = (const float*)d_in[5];
  const float* fgn_b   = (const float*)d_in[6];
  const float* root_w  = (const float*)d_in[7];
  const float* conv_b  = (const float*)d_in[8];
  const float* dense_w = (const float*)d_in[9];
  const float* dense_b = (const float*)d_in[10];
  float* out = (float*)d_out;

  float* agg    = (float*)d_ws;                     // [N_NODES x 32] = 12.8 MB
  float* pooled = agg + (size_t)N_NODES * CHANNELS; // [64 x 32]
  const int zeroN = N_NODES * CHANNELS + N_GRAPHS * CHANNELS;

  zero_ws_kernel<<<512, 256, 0, stream>>>(agg, zeroN);

  // 8192 waves grid-striding over 62500 16-edge tiles
  ecc_edge_wmma_kernel<<<1024, 256, 0, stream>>>(x, src, dst, e, fgn_w, fgn_b, agg);

  const int nodesPerWave = 64;                      // 1563 waves
  const int nodeWaves  = (N_NODES + nodesPerWave - 1) / nodesPerWave;
  const int nodeBlocks = (nodeWaves + 7) / 8;
  node_relu_pool_kernel<<<nodeBlocks, 256, 0, stream>>>(x, agg, root_w, conv_b,
                                                        gidx, pooled, nodesPerWave);

  head_softmax_kernel<<<1, 64, 0, stream>>>(pooled, dense_w, dense_b, out);
}